// GC_TG_LSTM_flu_55224689492626
// MI455X (gfx1250) — compile-verified
//
#include <hip/hip_runtime.h>
#include <hip/hip_bf16.h>
#include <math.h>

// ---- problem constants (from reference) ----
#define BB   8
#define TT   168
#define BT   1344          // B*T
#define NN   425
#define EE   128
#define II   128
#define HH   256
#define F1D  10
#define F2D  32
#define KGFC 13600         // N*F2

typedef __attribute__((ext_vector_type(2))) float v2f;
typedef __attribute__((ext_vector_type(8))) float v8f;

// =====================================================================
// Kernel 1: two-layer GCN per (b,t) graph, fully LDS-resident.
// grid = BT blocks, 256 threads. Writes x2[bt][n*32+j] (relu'd) to ws.
// =====================================================================
__global__ void gcn_kernel(const float* __restrict__ node_feat,
                           const int*   __restrict__ esrc_g,
                           const int*   __restrict__ edst_g,
                           const float* __restrict__ ew_g,
                           const float* __restrict__ Wg1,  // [1,10]
                           const float* __restrict__ bg1,  // [10]
                           const float* __restrict__ Wg2,  // [10,32]
                           const float* __restrict__ bg2,  // [32]
                           float* __restrict__ x2buf)      // [BT,13600]
{
    __shared__ float deg[NN];
    __shared__ float dinv[NN];
    __shared__ int   es[EE];
    __shared__ int   ed[EE];
    __shared__ float cf[EE];
    __shared__ float h1[NN][F1D];   // h after x@Wg1; later holds x1
    __shared__ float a1[NN][F1D];   // layer-1 aggregate
    __shared__ float h2[NN][4];     // layer-2 feature chunk
    __shared__ float a2[NN][4];     // layer-2 aggregate chunk

    const int bt  = blockIdx.x;
    const int tid = threadIdx.x;
    const size_t ebase = (size_t)bt * EE;
    const size_t nbase = (size_t)bt * NN;

    for (int n = tid; n < NN; n += 256) deg[n] = 0.f;
    for (int e = tid; e < EE; e += 256) {
        es[e] = esrc_g[ebase + e];
        ed[e] = edst_g[ebase + e];
    }
    __syncthreads();
    for (int e = tid; e < EE; e += 256)
        atomicAdd(&deg[ed[e]], ew_g[ebase + e]);
    __syncthreads();
    // self-loop weight 1 folded in; deg >= 1 always
    for (int n = tid; n < NN; n += 256) dinv[n] = rsqrtf(deg[n] + 1.0f);
    for (int idx = tid; idx < NN * F1D; idx += 256) {
        int n = idx / F1D, f = idx % F1D;
        h1[n][f] = node_feat[nbase + n] * Wg1[f];
    }
    __syncthreads();
    // layer-1 self term + edge norm coefficients
    for (int idx = tid; idx < NN * F1D; idx += 256) {
        int n = idx / F1D, f = idx % F1D;
        a1[n][f] = dinv[n] * dinv[n] * h1[n][f];
    }
    for (int e = tid; e < EE; e += 256)
        cf[e] = dinv[es[e]] * ew_g[ebase + e] * dinv[ed[e]];
    __syncthreads();
    for (int idx = tid; idx < EE * F1D; idx += 256) {
        int e = idx / F1D, f = idx % F1D;
        atomicAdd(&a1[ed[e]][f], cf[e] * h1[es[e]][f]);
    }
    __syncthreads();
    // x1 = relu(a1 + bg1)  (stored back into h1)
    for (int idx = tid; idx < NN * F1D; idx += 256) {
        int n = idx / F1D, f = idx % F1D;
        float v = a1[n][f] + bg1[f];
        h1[n][f] = v > 0.f ? v : 0.f;
    }
    __syncthreads();
    // layer 2: 8 chunks of 4 output features
    for (int fc = 0; fc < 8; ++fc) {
        for (int idx = tid; idx < NN * 4; idx += 256) {
            int n = idx >> 2, q = idx & 3;
            int j = fc * 4 + q;
            float s = 0.f;
#pragma unroll
            for (int f = 0; f < F1D; ++f) s += h1[n][f] * Wg2[f * F2D + j];
            h2[n][q] = s;
        }
        __syncthreads();
        for (int idx = tid; idx < NN * 4; idx += 256) {
            int n = idx >> 2, q = idx & 3;
            a2[n][q] = dinv[n] * dinv[n] * h2[n][q];
        }
        __syncthreads();
        for (int idx = tid; idx < EE * 4; idx += 256) {
            int e = idx >> 2, q = idx & 3;
            atomicAdd(&a2[ed[e]][q], cf[e] * h2[es[e]][q]);
        }
        __syncthreads();
        // vectorized relu+bias store: one float4 (global_store_b128) per node
        for (int n = tid; n < NN; n += 256) {
            float4 v;
            float t0 = a2[n][0] + bg2[fc * 4 + 0];
            float t1 = a2[n][1] + bg2[fc * 4 + 1];
            float t2 = a2[n][2] + bg2[fc * 4 + 2];
            float t3 = a2[n][3] + bg2[fc * 4 + 3];
            v.x = t0 > 0.f ? t0 : 0.f;
            v.y = t1 > 0.f ? t1 : 0.f;
            v.z = t2 > 0.f ? t2 : 0.f;
            v.w = t3 > 0.f ? t3 : 0.f;
            *(float4*)&x2buf[(size_t)bt * KGFC + n * F2D + fc * 4] = v;
        }
        __syncthreads();
    }
}

// =====================================================================
// Kernel 2: fp32 WMMA GEMM: D = relu(A[M,K] * B[K,N] + bias[N])
// 256 threads = 8 waves per block. Block owns one 16-row M strip; the
// A K-chunk is staged into LDS ONCE per block and shared by all waves
// (8x less A traffic than per-tile blocks). Each wave owns one 16x16
// N-tile: stages its B chunk, runs 8x v_wmma_f32_16x16x4_f32 per
// 32-wide K-chunk. float4 (b128) staging on the aligned fast path,
// predicated scalar fallback for ragged M/N/K edges.
// grid = (ceil(M/16), ceil(N/128)), block = 256.
// =====================================================================
__global__ void __launch_bounds__(256)
wmma_gemm_bias_relu(const float* __restrict__ A, int lda,
                    const float* __restrict__ Bw, int ldb,
                    const float* __restrict__ bias,
                    float* __restrict__ D, int ldd,
                    int M, int N, int K)
{
    __shared__ float As[16][36];        // stride 36: b128-aligned rows,
                                        // conflict-free frag reads
    __shared__ float Bs[8][32][16];     // per-wave B chunk

    const int tid  = threadIdx.x;
    const int lane = tid & 31;
    const int wave = tid >> 5;          // 0..7
    const int sub  = lane >> 4;
    const int l15  = lane & 15;

    const int m0  = blockIdx.x * 16;
    const int n0w = blockIdx.y * 128 + wave * 16;   // this wave's N-tile
    const bool tileActive = (n0w < N);

    v8f acc = {};
    const int kchunks = (K + 31) / 32;
    const bool aVecOK = ((lda & 3) == 0) && (m0 + 16 <= M);
    const bool bVecOK = ((ldb & 3) == 0) && (n0w + 16 <= N);

    for (int ch = 0; ch < kchunks; ++ch) {
        const int kc = ch * 32;
        const bool fullK = (kc + 32 <= K);

        // ---- stage A chunk (block-cooperative, shared by all 8 waves) ----
        if (aVecOK && fullK) {
            // 16 rows x 8 float4 = 128 items; threads 0..127
            if (tid < 128) {
                int r = tid >> 3, vq = tid & 7;
                float4 v = *(const float4*)&A[(size_t)(m0 + r) * lda + kc + vq * 4];
                *(float4*)&As[r][vq * 4] = v;
                // prefetch next K-chunk of the A stream (global_prefetch_b8)
                if (kc + 32 < K)
                    __builtin_prefetch(&A[(size_t)(m0 + r) * lda + kc + 32 + vq * 4], 0, 0);
            }
        } else {
            // 16 rows x 32 cols = 512 items; 256 threads -> 2 each
            for (int idx = tid; idx < 512; idx += 256) {
                int r = idx >> 5, k = kc + (idx & 31);
                int m = m0 + r;
                As[r][idx & 31] = (m < M && k < K) ? A[(size_t)m * lda + k] : 0.f;
            }
        }

        // ---- stage this wave's B chunk ----
        if (tileActive) {
            if (bVecOK && fullK) {
                // 32 rows x 4 float4 = 128 items; 32 lanes -> 4 each
#pragma unroll
                for (int j = 0; j < 4; ++j) {
                    int item = lane + 32 * j;
                    int kk = item >> 2, q = item & 3;
                    float4 v = *(const float4*)&Bw[(size_t)(kc + kk) * ldb + n0w + q * 4];
                    *(float4*)&Bs[wave][kk][q * 4] = v;
                }
            } else {
#pragma unroll
                for (int r = 0; r < 16; ++r) {
                    int kk = 2 * r + sub;
                    int k = kc + kk, n = n0w + l15;
                    Bs[wave][kk][l15] =
                        (k < K && n < N) ? Bw[(size_t)k * ldb + n] : 0.f;
                }
            }
        }
        __syncthreads();

        // ---- 8x WMMA over the 32-wide K chunk ----
        if (tileActive) {
#pragma unroll
            for (int i = 0; i < 8; ++i) {
                // A 16x4 frag: lane -> m=l15, vgpr v -> k = 4i + 2*sub + v
                v2f a, b;
                a[0] = As[l15][4 * i + 2 * sub + 0];
                a[1] = As[l15][4 * i + 2 * sub + 1];
                // B 4x16 frag: lane -> n=l15, vgpr v -> k = 4i + 2*sub + v
                b[0] = Bs[wave][4 * i + 2 * sub + 0][l15];
                b[1] = Bs[wave][4 * i + 2 * sub + 1][l15];
                acc = __builtin_amdgcn_wmma_f32_16x16x4_f32(
                    false, a, false, b, (short)0, acc, false, false);
            }
        }
        __syncthreads();
    }

    // ---- epilogue: C/D layout -> row m = r + 8*sub, col n = l15 ----
    if (tileActive) {
        const int n = n0w + l15;
        const float bv = (n < N) ? bias[n] : 0.f;
#pragma unroll
        for (int r = 0; r < 8; ++r) {
            int m = m0 + r + 8 * sub;
            if (m < M && n < N) {
                float v = acc[r] + bv;
                D[(size_t)m * ldd + n] = v > 0.f ? v : 0.f;
            }
        }
    }
}

// =====================================================================
// Kernel 3: persistent single-workgroup LSTM scan (the sequential path).
// 1024 threads = 32 wave32 waves. h/c/G in LDS. Per step:
//   G[8,1024] = [x_t | h] (8x384, zero-padded to 16 rows) x W^T via WMMA
//   then fused sigmoid/tanh gate update. Emits hr = relu(h_T).
// =====================================================================
__global__ void __launch_bounds__(1024)
lstm_kernel(const float* __restrict__ li,    // [BT, I] (b-major rows b*T+t)
            const float* __restrict__ h0,    // [B,H]
            const float* __restrict__ c0,    // [B,H]
            const float* __restrict__ Wih,   // [4H, I]
            const float* __restrict__ Whh,   // [4H, H]
            const float* __restrict__ bih,   // [4H]
            const float* __restrict__ bhh,   // [4H]
            float* __restrict__ hr)          // [B,H] relu'd final h
{
    __shared__ float h[BB][HH];          // 8 KB
    __shared__ float c[BB][HH];          // 8 KB
    __shared__ float xh[BB][II + HH];    // 12 KB  [x_t | h]
    __shared__ float G[BB][4 * HH];      // 32 KB gate pre-activations

    const int tid  = threadIdx.x;
    const int lane = tid & 31;
    const int wave = tid >> 5;           // 0..31
    const int sub  = lane >> 4;
    const int l15  = lane & 15;

    for (int idx = tid; idx < BB * HH; idx += 1024) {
        h[idx >> 8][idx & 255] = h0[idx];
        c[idx >> 8][idx & 255] = c0[idx];
    }
    __syncthreads();

    for (int t = 0; t < TT; ++t) {
        // stage [x_t | h] into LDS
        for (int idx = tid; idx < BB * II; idx += 1024) {
            int b = idx >> 7, k = idx & 127;
            xh[b][k] = li[((size_t)b * TT + t) * II + k];
        }
        for (int idx = tid; idx < BB * HH; idx += 1024) {
            int b = idx >> 8, k = idx & 255;
            xh[b][II + k] = h[b][k];
        }
        __syncthreads();

        // G = xh @ [Wih | Whh]^T : 64 N-tiles of 16, 2 per wave, K=384
        for (int nt = wave * 2; nt < wave * 2 + 2; ++nt) {
            const int n = nt * 16 + l15;      // gate column this lane owns
            v8f acc = {};
#pragma unroll 4
            for (int i = 0; i < 96; ++i) {
                const int kb = 4 * i + 2 * sub;   // even, never straddles 128
                v2f a, b;
                a[0] = (l15 < BB) ? xh[l15][kb] : 0.f;     // rows 8..15 pad 0
                a[1] = (l15 < BB) ? xh[l15][kb + 1] : 0.f;
                if (kb < II) {
                    b[0] = Wih[(size_t)n * II + kb];
                    b[1] = Wih[(size_t)n * II + kb + 1];
                } else {
                    b[0] = Whh[(size_t)n * HH + (kb - II)];
                    b[1] = Whh[(size_t)n * HH + (kb - II) + 1];
                }
                acc = __builtin_amdgcn_wmma_f32_16x16x4_f32(
                    false, a, false, b, (short)0, acc, false, false);
            }
            if (sub == 0) {                  // rows 0..7 are the real batch
#pragma unroll
                for (int r = 0; r < 8; ++r) G[r][n] = acc[r];
            }
        }
        __syncthreads();

        // fused gate update (order i,f,g,o)
        for (int idx = tid; idx < BB * HH; idx += 1024) {
            int b = idx >> 8, j = idx & 255;
            float gi = G[b][j]           + bih[j]           + bhh[j];
            float gf = G[b][HH + j]      + bih[HH + j]      + bhh[HH + j];
            float gg = G[b][2 * HH + j]  + bih[2 * HH + j]  + bhh[2 * HH + j];
            float go = G[b][3 * HH + j]  + bih[3 * HH + j]  + bhh[3 * HH + j];
            float si = 1.f / (1.f + expf(-gi));
            float sf = 1.f / (1.f + expf(-gf));
            float so = 1.f / (1.f + expf(-go));
            float cn = sf * c[b][j] + si * tanhf(gg);
            c[b][j] = cn;
            h[b][j] = so * tanhf(cn);
        }
        __syncthreads();
    }

    for (int idx = tid; idx < BB * HH; idx += 1024) {
        float v = h[idx >> 8][idx & 255];
        hr[idx] = v > 0.f ? v : 0.f;
    }
}

// =====================================================================
// Host-side orchestration
// =====================================================================
extern "C" void kernel_launch(void* const* d_in, const int* in_sizes, int n_in,
                              void* d_out, int out_size, void* d_ws, size_t ws_size,
                              hipStream_t stream) {
    const float* node_feat = (const float*)d_in[0];
    const int*   esrc      = (const int*)  d_in[1];
    const int*   edst      = (const int*)  d_in[2];
    const float* ew        = (const float*)d_in[3];
    const float* infect    = (const float*)d_in[4];
    const float* temp      = (const float*)d_in[5];
    const float* h0        = (const float*)d_in[6];
    const float* c0        = (const float*)d_in[7];
    const float* Wg1  = (const float*)d_in[8];
    const float* bg1  = (const float*)d_in[9];
    const float* Wg2  = (const float*)d_in[10];
    const float* bg2  = (const float*)d_in[11];
    const float* Wgfc = (const float*)d_in[12];
    const float* bgfc = (const float*)d_in[13];
    const float* Wi1  = (const float*)d_in[14];
    const float* bi1  = (const float*)d_in[15];
    const float* Wi2  = (const float*)d_in[16];
    const float* bi2  = (const float*)d_in[17];
    const float* Wi3  = (const float*)d_in[18];
    const float* bi3  = (const float*)d_in[19];
    const float* Wi4  = (const float*)d_in[20];
    const float* bi4  = (const float*)d_in[21];
    const float* Wt1  = (const float*)d_in[22];
    const float* bt1  = (const float*)d_in[23];
    const float* Wt2  = (const float*)d_in[24];
    const float* bt2  = (const float*)d_in[25];
    const float* Wcat = (const float*)d_in[26];
    const float* bcat = (const float*)d_in[27];
    const float* Wih  = (const float*)d_in[28];
    const float* Whh  = (const float*)d_in[29];
    const float* bih  = (const float*)d_in[30];
    const float* bhh  = (const float*)d_in[31];
    const float* Wfc  = (const float*)d_in[32];
    const float* bfc  = (const float*)d_in[33];

    // workspace carve-up (floats)
    float* ws    = (float*)d_ws;
    float* x2buf = ws;                                   // [1344, 13600]
    float* cat   = x2buf + (size_t)BT * KGFC;            // [1344, 384]
    float* li    = cat   + (size_t)BT * 384;             // [1344, 128]
    float* inf1  = li    + (size_t)BT * II;              // [1344, 128]
    float* inf2  = inf1  + (size_t)BT * 128;             // [1344, 128]
    float* inf3  = inf2  + (size_t)BT * 128;             // [1344, 64]
    float* tp1   = inf3  + (size_t)BT * 64;              // [1344, 64]
    float* hr    = tp1   + (size_t)BT * 64;              // [8, 256]

    const dim3 b256(256, 1, 1);

    // Phase A: GCN (both layers) -> x2
    gcn_kernel<<<BT, 256, 0, stream>>>(node_feat, esrc, edst, ew,
                                       Wg1, bg1, Wg2, bg2, x2buf);

    // Phase B: branch GEMMs (all relu+bias), concat assembled in-place
    wmma_gemm_bias_relu<<<dim3(BT / 16, 1), b256, 0, stream>>>(
        x2buf, KGFC, Wgfc, II, bgfc, cat + 0, 384, BT, II, KGFC);      // graph branch
    wmma_gemm_bias_relu<<<dim3(BT / 16, 1), b256, 0, stream>>>(
        infect, 25, Wi1, 128, bi1, inf1, 128, BT, 128, 25);
    wmma_gemm_bias_relu<<<dim3(BT / 16, 1), b256, 0, stream>>>(
        inf1, 128, Wi2, 128, bi2, inf2, 128, BT, 128, 128);
    wmma_gemm_bias_relu<<<dim3(BT / 16, 1), b256, 0, stream>>>(
        inf2, 128, Wi3, 64, bi3, inf3, 64, BT, 64, 128);
    wmma_gemm_bias_relu<<<dim3(BT / 16, 1), b256, 0, stream>>>(
        inf3, 64, Wi4, 128, bi4, cat + 128, 384, BT, 128, 64);         // infect branch
    wmma_gemm_bias_relu<<<dim3(BT / 16, 1), b256, 0, stream>>>(
        temp, 13, Wt1, 64, bt1, tp1, 64, BT, 64, 13);
    wmma_gemm_bias_relu<<<dim3(BT / 16, 1), b256, 0, stream>>>(
        tp1, 64, Wt2, 128, bt2, cat + 256, 384, BT, 128, 64);          // temp branch
    wmma_gemm_bias_relu<<<dim3(BT / 16, 1), b256, 0, stream>>>(
        cat, 384, Wcat, 128, bcat, li, 128, BT, 128, 384);             // concat fc

    // Phase C: sequential LSTM scan, single persistent workgroup
    lstm_kernel<<<1, 1024, 0, stream>>>(li, h0, c0, Wih, Whh, bih, bhh, hr);

    // Phase D: final FC straight into d_out ([8,850] flat == user|infect split)
    wmma_gemm_bias_relu<<<dim3(1, (850 + 127) / 128), b256, 0, stream>>>(
        hr, HH, Wfc, 850, bfc, (float*)d_out, 850, BB, 850, HH);
}